// TS_Stddev_75703093559346
// MI455X (gfx1250) — compile-verified
//
#include <hip/hip_runtime.h>

// Sliding-window stddev (window=10, stride=1) over time dim of x[64,4096,256] f32.
// Memory-bound: ~0.54 GB traffic -> ~23us floor at 23.3 TB/s; AI ~1.5 FLOP/B, so
// matrix ops are useless — the win is the CDNA5 async data-movement path:
//   global_load_async_to_lds_b128 (ASYNCcnt-tracked DMA) stages rows 24 ahead into
//   a 32-slot LDS ring; compute reads newest/oldest window rows from LDS; no
//   barriers needed because every LDS dependency is wave-local (each lane copies
//   and reads only its own 16B column); non-temporal b128 stores for the output.

typedef float v4f __attribute__((ext_vector_type(4)));

#define DAYS    10
#define T_DIM   4096
#define C_DIM   256
#define B_DIM   64
#define NW      (T_DIM - DAYS + 1)            // 4087 windows
#define TTILE   130                           // windows per block
#define NTILES  ((NW + TTILE - 1) / TTILE)    // 32
#define C4      (C_DIM / 4)                   // 64 float4 columns per row
#define SLOTS   32                            // LDS ring depth (rows)
#define AHEAD   24                            // issue distance (rows)
#define ROW_B   (C4 * 16)                     // 1024 bytes per staged row

__global__ __launch_bounds__(C4) void ts_stddev_async_kernel(const float* __restrict__ x,
                                                             float* __restrict__ out) {
  __shared__ v4f buf[SLOTS * C4];             // 32 KB circular stage

  const int tid = threadIdx.x;                // one float4 (4 channels) per thread
  const int b   = blockIdx.y;
  const int w0  = blockIdx.x * TTILE;

  const v4f* __restrict__ px = (const v4f*)x + (size_t)b * T_DIM * C4 + tid;  // row t = px[t*C4]
  v4f*       __restrict__ po = (v4f*)out + ((size_t)b * NW + w0) * C4 + tid;

  // This lane's LDS column base (byte offset; slot s lives at lbase + s*ROW_B).
  const unsigned lbase = (unsigned)(size_t)(&buf[tid]);

  // ---- Prologue: issue rows 0..AHEAD-1 (times w0..w0+23, always in range since
  //      w0 <= 4030 and 4030+23 < 4096). ASYNCcnt <= 24.
  for (int r = 0; r < AHEAD; ++r) {
    const v4f* gp  = px + (size_t)(w0 + r) * C4;
    unsigned   lds = lbase + (unsigned)(r & (SLOTS - 1)) * ROW_B;
    asm volatile("global_load_async_to_lds_b128 %0, %1, off"
                 :: "v"(lds), "v"(gp) : "memory");
  }
  // Loads complete in order: <=15 outstanding of 24 issued => rows 0..8 resident.
  asm volatile("s_wait_asynccnt 15" ::: "memory");

  v4f s = {0.f, 0.f, 0.f, 0.f};
  v4f q = {0.f, 0.f, 0.f, 0.f};
#pragma unroll
  for (int r = 0; r < DAYS - 1; ++r) {
    v4f v = buf[r * C4 + tid];
    s += v;
    q += v * v;
  }

  const int kend = (NW - w0 < TTILE) ? (NW - w0) : TTILE;   // valid windows in tile
#pragma unroll 2
  for (int k = 0; k < kend; ++k) {
    // Issue row k+AHEAD. Clamp the time index in-bounds: clamped rows land only in
    // slots that are never consumed (any slot later read as "new" at k'=k+AHEAD-9
    // or "old" at k'=k+AHEAD implies w0+k+AHEAD <= 4095, i.e. no clamp happened).
    int t_issue = w0 + k + AHEAD;
    if (t_issue > T_DIM - 1) t_issue = T_DIM - 1;
    {
      const v4f* gp  = px + (size_t)t_issue * C4;
      unsigned   lds = lbase + (unsigned)((k + AHEAD) & (SLOTS - 1)) * ROW_B;
      asm volatile("global_load_async_to_lds_b128 %0, %1, off"
                   :: "v"(lds), "v"(gp) : "memory");
    }
    // After issuing row k+AHEAD: <=15 outstanding => rows <= k+9 resident.
    asm volatile("s_wait_asynccnt 15" ::: "memory");

    v4f vn = buf[((k + DAYS - 1) & (SLOTS - 1)) * C4 + tid];  // newest row of window k
    s += vn;
    q += vn * vn;

    // Unbiased variance: (sumsq - sum^2/n) / (n-1)
    v4f var = (q - s * s * (1.0f / DAYS)) * (1.0f / (DAYS - 1));
    v4f sd;
#pragma unroll
    for (int i = 0; i < 4; ++i)
      sd[i] = __builtin_sqrtf(__builtin_fmaxf(var[i], 0.0f));
    __builtin_nontemporal_store(sd, po + (size_t)k * C4);

    v4f vo = buf[(k & (SLOTS - 1)) * C4 + tid];               // row leaving the window
    s -= vo;                                                   // (slot overwritten only at
    q -= vo * vo;                                              //  iteration k+8 — safe)
  }
  // S_ENDPGM performs an implicit wait-idle for any still-outstanding async loads.
}

extern "C" void kernel_launch(void* const* d_in, const int* in_sizes, int n_in,
                              void* d_out, int out_size, void* d_ws, size_t ws_size,
                              hipStream_t stream) {
  const float* x = (const float*)d_in[0];
  float* out = (float*)d_out;

  dim3 grid(NTILES, B_DIM, 1);
  dim3 block(C4, 1, 1);
  ts_stddev_async_kernel<<<grid, block, 0, stream>>>(x, out);

  (void)in_sizes; (void)n_in; (void)out_size; (void)d_ws; (void)ws_size;
}